// DecorrelatedBatchNorm_78503412236531
// MI455X (gfx1250) — compile-verified
//
#include <hip/hip_runtime.h>

typedef __attribute__((ext_vector_type(2))) float v2f;
typedef __attribute__((ext_vector_type(8))) float v8f;

#define NUM_FEATURES 256
#define GROUP_SIZE   32
#define NUM_GROUPS   8
#define BATCH        32
#define HW           4096          // 64*64
#define NCOLS        (BATCH * HW)  // 131072 columns per group
#define EPS          1e-5f
#define NS_ITERS     32

// workspace layout (in floats)
#define WS_GRAM   0
#define WS_SUMS   (NUM_GROUPS * 32 * 32)            // 8192
#define WS_WM     (WS_SUMS + NUM_GROUPS * 32)       // 8448
#define WS_SHIFT  (WS_WM + NUM_GROUPS * 32 * 32)    // 16640

// ---------------------------------------------------------------------------
// Kernel 1: per-group Gram matrix (X * X^T, 32x32) + per-channel sums.
// Each wave owns 256 contiguous columns inside one image -> base address is
// computed once and the inner loop is immediate-offset global_load_b64 only.
// One float2 per lane doubles as both the A (16x4) and B (4x16) fragment of
// V_WMMA_F32_16X16X4_F32 (lane layouts coincide for Gram).
// ---------------------------------------------------------------------------
__global__ void __launch_bounds__(256) dbn_gram(const float* __restrict__ x,
                                                float* __restrict__ ws) {
    const int g      = blockIdx.y;
    const int lane   = threadIdx.x & 31;
    const int wave   = threadIdx.x >> 5;
    const int lane16 = lane & 15;
    const int half   = lane >> 4;
    const int wid    = blockIdx.x * 8 + wave;   // 0..511 within this group

    float* gram = ws + WS_GRAM + g * 1024;
    float* sums = ws + WS_SUMS + g * 32;

    // 512 waves/group, 256 contiguous columns each (16 waves per image)
    const int b  = wid >> 4;
    const int s0 = (wid & 15) * 256;

    const float* pLo = x + (size_t)(b * NUM_FEATURES + g * GROUP_SIZE + lane16) * HW
                         + (size_t)(s0 + 2 * half);
    const float* pHi = pLo + (size_t)16 * HW;

    v8f a00 = {}, a01 = {}, a10 = {}, a11 = {};
    float sLo = 0.f, sHi = 0.f;

    #pragma unroll 4
    for (int i = 0; i < 64; ++i) {              // 64 chunks of 4 columns
        v2f lo = *(const v2f*)(pLo + i * 4);
        v2f hi = *(const v2f*)(pHi + i * 4);

        a00 = __builtin_amdgcn_wmma_f32_16x16x4_f32(false, lo, false, lo, (short)0, a00, false, false);
        a01 = __builtin_amdgcn_wmma_f32_16x16x4_f32(false, lo, false, hi, (short)0, a01, false, false);
        a10 = __builtin_amdgcn_wmma_f32_16x16x4_f32(false, hi, false, lo, (short)0, a10, false, false);
        a11 = __builtin_amdgcn_wmma_f32_16x16x4_f32(false, hi, false, hi, (short)0, a11, false, false);

        sLo += lo.x + lo.y;
        sHi += hi.x + hi.y;
    }

    // channel sums: lane L and lane L^16 both hold partials of channel (L&15)
    sLo += __shfl_xor(sLo, 16, 32);
    sHi += __shfl_xor(sHi, 16, 32);
    if (half == 0) {
        atomicAdd(&sums[lane16],      sLo);
        atomicAdd(&sums[lane16 + 16], sHi);
    }

    // D tile layout: lanes 0-15 vgpr j -> (M=j, N=lane); lanes 16-31 -> (M=j+8)
    #pragma unroll
    for (int j = 0; j < 8; ++j) {
        const int m = j + 8 * half;
        const int n = lane16;
        atomicAdd(&gram[(m)      * 32 + n],      a00[j]);
        atomicAdd(&gram[(m)      * 32 + n + 16], a01[j]);
        atomicAdd(&gram[(m + 16) * 32 + n],      a10[j]);
        atomicAdd(&gram[(m + 16) * 32 + n + 16], a11[j]);
    }
}

// ---------------------------------------------------------------------------
// Kernel 2: sigma = Gram - N*mean*mean^T + eps*I, then coupled Newton-Schulz
// iteration: Z -> (sigma/trace)^{-1/2}; wm = Z / sqrt(trace).
// One 32x32-thread block per group; all math in LDS. Also emits
// shift = wm @ mean for the fused apply epilogue.
// ---------------------------------------------------------------------------
__global__ void __launch_bounds__(1024) dbn_invsqrt(float* __restrict__ ws) {
    const int g = blockIdx.x;
    const int j = threadIdx.x & 31;
    const int i = threadIdx.x >> 5;

    __shared__ float Y[32][33], Z[32][33], T[32][33];
    __shared__ float tr;

    const float* gram = ws + WS_GRAM + g * 1024;
    const float* sums = ws + WS_SUMS + g * 32;
    float* wm    = ws + WS_WM    + g * 1024;
    float* shift = ws + WS_SHIFT + g * 32;

    const float invN = 1.0f / (float)NCOLS;
    const float mi = sums[i] * invN;
    const float mj = sums[j] * invN;
    float sij = gram[i * 32 + j] - (float)NCOLS * mi * mj + ((i == j) ? EPS : 0.f);

    if (threadIdx.x == 0) tr = 0.f;
    __syncthreads();
    if (i == j) atomicAdd(&tr, sij);
    __syncthreads();
    const float t = tr;

    Y[i][j] = sij / t;
    Z[i][j] = (i == j) ? 1.f : 0.f;

    for (int it = 0; it < NS_ITERS; ++it) {
        __syncthreads();
        float zy = 0.f;
        #pragma unroll
        for (int k = 0; k < 32; ++k) zy += Z[i][k] * Y[k][j];
        T[i][j] = ((i == j) ? 3.f : 0.f) - zy;
        __syncthreads();
        float yt = 0.f, tz = 0.f;
        #pragma unroll
        for (int k = 0; k < 32; ++k) { yt += Y[i][k] * T[k][j]; tz += T[i][k] * Z[k][j]; }
        __syncthreads();
        Y[i][j] = 0.5f * yt;
        Z[i][j] = 0.5f * tz;
    }
    __syncthreads();

    const float w = Z[i][j] * rsqrtf(t);
    wm[i * 32 + j] = w;

    T[i][j] = w * mj;              // reuse T for row-reduction of wm @ mean
    __syncthreads();
    if (j == 0) {
        float s = 0.f;
        #pragma unroll
        for (int k = 0; k < 32; ++k) s += T[i][k];
        shift[i] = s;
    }
}

// ---------------------------------------------------------------------------
// Kernel 3: out = (wm @ x - shift) * weight + bias via WMMA.
// A fragments (wm) live in registers, reused across all 8 column tiles.
// Epilogue folded to a single FMA per element; nontemporal stores keep the
// 134MB output from evicting x out of L2.
// ---------------------------------------------------------------------------
__global__ void __launch_bounds__(256) dbn_apply(const float* __restrict__ x,
                                                 const float* __restrict__ weight,
                                                 const float* __restrict__ bias,
                                                 const float* __restrict__ ws,
                                                 float* __restrict__ out) {
    const int g      = blockIdx.y;
    const int lane   = threadIdx.x & 31;
    const int wave   = threadIdx.x >> 5;
    const int lane16 = lane & 15;
    const int half   = lane >> 4;
    const int wid    = blockIdx.x * 8 + wave;   // 0..1023 within this group

    __shared__ float sw[32], sb[32], ssh[32];
    if (threadIdx.x < 32) {
        const int c = g * 32 + threadIdx.x;
        sw[threadIdx.x]  = weight[c];
        sb[threadIdx.x]  = bias[c];
        ssh[threadIdx.x] = ws[WS_SHIFT + g * 32 + threadIdx.x];
    }
    __syncthreads();

    // Preload wm A fragments: a0 = out rows 0-15, a1 = out rows 16-31.
    // Lane holds (M=lane16[+16], K=ks*4 + 2*half .. +1): contiguous float2.
    const float* wm = ws + WS_WM + g * 1024;
    v2f a0[8], a1[8];
    #pragma unroll
    for (int ks = 0; ks < 8; ++ks) {
        a0[ks] = *(const v2f*)(wm + lane16 * 32 + ks * 4 + 2 * half);
        a1[ks] = *(const v2f*)(wm + (lane16 + 16) * 32 + ks * 4 + 2 * half);
    }

    // Per-lane epilogue constants: v = acc * wA + cA  (folds shift/weight/bias)
    float wA[8], cA[8], wB[8], cB[8];
    #pragma unroll
    for (int jj = 0; jj < 8; ++jj) {
        const int m = jj + 8 * half;
        wA[jj] = sw[m];       cA[jj] = sb[m]      - ssh[m]      * sw[m];
        wB[jj] = sw[m + 16];  cB[jj] = sb[m + 16] - ssh[m + 16] * sw[m + 16];
    }

    // 1024 waves/group, 128 contiguous columns each (32 waves per image)
    const int b  = wid >> 5;
    const int s0 = (wid & 31) * 128;

    const float* px = x + (size_t)(b * NUM_FEATURES + g * GROUP_SIZE) * HW
                        + (size_t)(s0 + lane16) + (size_t)(2 * half) * HW;
    float* po = out + (size_t)(b * NUM_FEATURES + g * GROUP_SIZE) * HW
                    + (size_t)(s0 + lane16) + (size_t)(8 * half) * HW;

    for (int t = 0; t < 8; ++t) {               // 8 tiles of 16 columns
        const int co = t * 16;
        v8f acc0 = {}, acc1 = {};
        #pragma unroll
        for (int ks = 0; ks < 8; ++ks) {
            v2f bf;
            bf.x = px[(size_t)(ks * 4) * HW + co];          // K = ks*4 + 2*half
            bf.y = px[(size_t)(ks * 4 + 1) * HW + co];      // K = +1
            acc0 = __builtin_amdgcn_wmma_f32_16x16x4_f32(false, a0[ks], false, bf, (short)0, acc0, false, false);
            acc1 = __builtin_amdgcn_wmma_f32_16x16x4_f32(false, a1[ks], false, bf, (short)0, acc1, false, false);
        }
        #pragma unroll
        for (int jj = 0; jj < 8; ++jj) {        // m = jj + 8*half folded into po
            __builtin_nontemporal_store(acc0[jj] * wA[jj] + cA[jj],
                                        po + (size_t)jj * HW + co);
            __builtin_nontemporal_store(acc1[jj] * wB[jj] + cB[jj],
                                        po + (size_t)(jj + 16) * HW + co);
        }
    }
}

// ---------------------------------------------------------------------------
extern "C" void kernel_launch(void* const* d_in, const int* in_sizes, int n_in,
                              void* d_out, int out_size, void* d_ws, size_t ws_size,
                              hipStream_t stream) {
    const float* x      = (const float*)d_in[0];
    const float* weight = (const float*)d_in[1];
    const float* bias   = (const float*)d_in[2];
    float* out = (float*)d_out;
    float* ws  = (float*)d_ws;

    // zero the gram + sums accumulators (graph-capture-safe)
    hipMemsetAsync(ws, 0,
                   (size_t)(WS_SUMS + NUM_GROUPS * 32) * sizeof(float), stream);

    dbn_gram   <<<dim3(64,  NUM_GROUPS), 256,  0, stream>>>(x, ws);
    dbn_invsqrt<<<dim3(NUM_GROUPS),      1024, 0, stream>>>(ws);
    dbn_apply  <<<dim3(128, NUM_GROUPS), 256,  0, stream>>>(x, weight, bias, ws, out);
}